// SkipGramModel_20856361189956
// MI455X (gfx1250) — compile-verified
//
#include <hip/hip_runtime.h>
#include <math.h>

#define DIM 128
#define LR_C 0.025f
#define LAP_C 0.01f
#define NEGW_C 1.0f

typedef float v2f __attribute__((ext_vector_type(2)));
typedef float v8f __attribute__((ext_vector_type(8)));

// fast_sigmoid: exact reproduction of the reference's 1202-entry table lookup,
// computed analytically (table[i] = sigmoid(-6.01 + 0.01*i), t[0]=0, t[1201]=1).
__device__ __forceinline__ float fast_sigmoid(float score) {
    float s = fminf(6.0f, fmaxf(-6.0f, score));
    int idx = (int)floorf((s + 6.01f) * 100.0f);
    if (idx <= 0) return 0.0f;
    if (idx >= 1201) return 1.0f;
    float x = -6.01f + 0.01f * (float)idx;
    return 1.0f / (1.0f + __expf(-x));
}

// Batched row-dot for 16 (u,v) row pairs per wave using V_WMMA_F32_16X16X4_F32.
// Lane L owns row m = L%16; h = L/16 selects the K sub-pair.
// A layout (ISA 7.12.2): a[v] holds U[m][K = v + 2h (+4s)] -> contiguous float2.
// B = V^T: b[v] holds B[v+2h][m] = V[m][v + 2h (+4s)]   -> same contiguous float2.
// Result D = U x V^T ; diagonal D[m][m] = dot(U_m, V_m).
// Returns dot for row m to *all* lanes with that m (via shuffle broadcast).
__device__ __forceinline__ float wmma_rowdot16(const float* __restrict__ pu,
                                               const float* __restrict__ pv,
                                               int lane) {
    const int h = lane >> 4;
    v8f acc = {0.f, 0.f, 0.f, 0.f, 0.f, 0.f, 0.f, 0.f};
#pragma unroll
    for (int s = 0; s < DIM / 4; ++s) {
        v2f a = *(const v2f*)(pu + 4 * s + 2 * h);
        v2f b = *(const v2f*)(pv + 4 * s + 2 * h);
        acc = __builtin_amdgcn_wmma_f32_16x16x4_f32(
            /*neg_a=*/false, a, /*neg_b=*/false, b,
            /*c_mod=*/(short)0, acc, /*reuse_a=*/false, /*reuse_b=*/false);
    }
    // C/D layout: c[v] at lane L is element (M = v + 8*(L/16), N = L%16).
    // Diagonal M==N lives in lanes 0..7 (rows 0..7, element c[lane]) and
    // lanes 24..31 (rows 8..15, element c[lane-24]).
    int dsel = (lane < 8) ? lane : (lane - 24);
    float sel = acc[0];
#pragma unroll
    for (int t = 1; t < 8; ++t) sel = (dsel == t) ? acc[t] : sel;
    int m = lane & 15;
    int src = (m < 8) ? m : (m + 16);
    return __shfl(sel, src);
}

// ---------------- Phase 1: positive-pair gradients (initializes gu/gv) -------
__global__ void k_pos(const float* __restrict__ w, const int* __restrict__ edges,
                      float* __restrict__ gu, float* __restrict__ gv, int batch) {
    const int lane = threadIdx.x & 31;
    const int wv   = threadIdx.x >> 5;
    const int ebase = (blockIdx.x * 8 + wv) * 16;
    if (ebase >= batch) return;  // wave-uniform: EXEC stays full for WMMA

    const int m = lane & 15;
    const int e = ebase + m;
    const int ui = edges[2 * e + 0];
    const int vi = edges[2 * e + 1];
    const float* pu = w + (size_t)ui * DIM;
    const float* pv = w + (size_t)vi * DIM;

    float score = wmma_rowdot16(pu, pv, lane);
    float coef = 1.0f - fast_sigmoid(score);

    for (int r = 0; r < 16; ++r) {
        float cr  = __shfl(coef, r);
        int   uir = __shfl(ui, r);
        int   vir = __shfl(vi, r);
        const float4 u4 = *(const float4*)(w + (size_t)uir * DIM + lane * 4);
        const float4 v4 = *(const float4*)(w + (size_t)vir * DIM + lane * 4);
        float4 g1, g2;
        g1.x = cr * v4.x + LAP_C * (v4.x - u4.x);
        g1.y = cr * v4.y + LAP_C * (v4.y - u4.y);
        g1.z = cr * v4.z + LAP_C * (v4.z - u4.z);
        g1.w = cr * v4.w + LAP_C * (v4.w - u4.w);
        g2.x = cr * u4.x + LAP_C * (u4.x - v4.x);
        g2.y = cr * u4.y + LAP_C * (u4.y - v4.y);
        g2.z = cr * u4.z + LAP_C * (u4.z - v4.z);
        g2.w = cr * u4.w + LAP_C * (u4.w - v4.w);
        *(float4*)(gu + (size_t)(ebase + r) * DIM + lane * 4) = g1;
        *(float4*)(gv + (size_t)(ebase + r) * DIM + lane * 4) = g2;
    }
}

// ---------------- Phase 2: negative samples (atomic accumulate into gu/gv) ---
__global__ void k_neg(const float* __restrict__ w, const int* __restrict__ edges,
                      const int* __restrict__ inu, const int* __restrict__ inv,
                      float* __restrict__ gu, float* __restrict__ gv, int nneg) {
    const int lane = threadIdx.x & 31;
    const int wv   = threadIdx.x >> 5;
    const int jbase = (blockIdx.x * 8 + wv) * 16;
    if (jbase >= nneg) return;

    const int m = lane & 15;
    const int j = jbase + m;
    const int iu = inu[j];
    const int iv = inv[j];
    const int un = edges[2 * iu + 0];   // emb_neg_u = weight[edges[idx_negu,0]]
    const int vn = edges[2 * iv + 1];   // emb_neg_v = weight[edges[idx_negv,1]]
    const float* pu = w + (size_t)un * DIM;
    const float* pv = w + (size_t)vn * DIM;

    float score = wmma_rowdot16(pu, pv, lane);
    float coef = -NEGW_C * fast_sigmoid(score);

    for (int r = 0; r < 16; ++r) {
        float cr  = __shfl(coef, r);
        int   iur = __shfl(iu, r);
        int   ivr = __shfl(iv, r);
        int   unr = __shfl(un, r);
        int   vnr = __shfl(vn, r);
        const float4 u4 = *(const float4*)(w + (size_t)unr * DIM + lane * 4);
        const float4 v4 = *(const float4*)(w + (size_t)vnr * DIM + lane * 4);
        float* du = gu + (size_t)iur * DIM + lane * 4;
        float* dv = gv + (size_t)ivr * DIM + lane * 4;
        atomicAdd(du + 0, cr * v4.x);
        atomicAdd(du + 1, cr * v4.y);
        atomicAdd(du + 2, cr * v4.z);
        atomicAdd(du + 3, cr * v4.w);
        atomicAdd(dv + 0, cr * u4.x);
        atomicAdd(dv + 1, cr * u4.y);
        atomicAdd(dv + 2, cr * u4.z);
        atomicAdd(dv + 3, cr * u4.w);
    }
}

// ---------------- Phase 3/5: scatter mean(grad^2) into state_sum -------------
__global__ void k_ss(const float* __restrict__ g, const int* __restrict__ edges,
                     int col, float* __restrict__ ss, int batch) {
    const int lane = threadIdx.x & 31;
    const int i = (int)((blockIdx.x * blockDim.x + threadIdx.x) >> 5); // 1 wave per edge
    if (i >= batch) return;
    const float4 g4 = *(const float4*)(g + (size_t)i * DIM + lane * 4);
    float s = g4.x * g4.x + g4.y * g4.y + g4.z * g4.z + g4.w * g4.w;
#pragma unroll
    for (int mask = 16; mask >= 1; mask >>= 1) s += __shfl_xor(s, mask);
    if (lane == 0) {
        int node = edges[2 * i + col];
        atomicAdd(ss + node, s * (1.0f / DIM));
    }
}

// ---------------- Phase 4/6: apply Adagrad-scaled update to weight -----------
__global__ void k_apply(const float* __restrict__ g, const int* __restrict__ edges,
                        int col, const float* __restrict__ ss,
                        float* __restrict__ wout, int batch) {
    const int lane = threadIdx.x & 31;
    const int i = (int)((blockIdx.x * blockDim.x + threadIdx.x) >> 5);
    if (i >= batch) return;
    const int node = edges[2 * i + col];
    const float stdv = sqrtf(ss[node]) + 1e-10f;
    const float k = LR_C / stdv;
    const float4 g4 = *(const float4*)(g + (size_t)i * DIM + lane * 4);
    float* d = wout + (size_t)node * DIM + lane * 4;
    atomicAdd(d + 0, k * g4.x);
    atomicAdd(d + 1, k * g4.y);
    atomicAdd(d + 2, k * g4.z);
    atomicAdd(d + 3, k * g4.w);
}

extern "C" void kernel_launch(void* const* d_in, const int* in_sizes, int n_in,
                              void* d_out, int out_size, void* d_ws, size_t ws_size,
                              hipStream_t stream) {
    const float* w    = (const float*)d_in[0];   // (EMB, 128) f32
    const float* ssin = (const float*)d_in[1];   // (EMB,)    f32
    const int* edges  = (const int*)d_in[2];     // (B, 2)    i32
    const int* inu    = (const int*)d_in[3];     // (5B,)     i32
    const int* inv    = (const int*)d_in[4];     // (5B,)     i32

    const int    batch = in_sizes[2] / 2;
    const int    nneg  = in_sizes[3];
    const size_t wlen  = (size_t)in_sizes[0];
    const size_t elen  = (size_t)in_sizes[1];

    float* wout  = (float*)d_out;
    float* ssout = wout + wlen;
    float* gu = (float*)d_ws;                    // (B,128) f32
    float* gv = gu + (size_t)batch * DIM;        // (B,128) f32

    // Phase 0: materialize outputs (bulk of the roofline: ~512 MB at 23.3 TB/s).
    hipMemcpyAsync(wout, w, wlen * sizeof(float), hipMemcpyDeviceToDevice, stream);
    hipMemcpyAsync(ssout, ssin, elen * sizeof(float), hipMemcpyDeviceToDevice, stream);

    const int b1 = (batch + 127) / 128;   // 8 waves/block, 16 edges/wave
    k_pos<<<b1, 256, 0, stream>>>(w, edges, gu, gv, batch);

    const int b2 = (nneg + 127) / 128;
    k_neg<<<b2, 256, 0, stream>>>(w, edges, inu, inv, gu, gv, nneg);

    const int b3 = (batch + 7) / 8;       // 1 wave per edge
    // u-step: scatter ss[u], then apply (reads ss post-u, pre-v — matches JAX order)
    k_ss   <<<b3, 256, 0, stream>>>(gu, edges, 0, ssout, batch);
    k_apply<<<b3, 256, 0, stream>>>(gu, edges, 0, ssout, wout, batch);
    // v-step
    k_ss   <<<b3, 256, 0, stream>>>(gv, edges, 1, ssout, batch);
    k_apply<<<b3, 256, 0, stream>>>(gv, edges, 1, ssout, wout, batch);
}